// BOCPD_GPTS_292057776458
// MI455X (gfx1250) — compile-verified
//
#include <hip/hip_runtime.h>
#include <hip/hip_bf16.h>

// CDNA5 / gfx1250: wave32, WMMA f32 16x16x4, 320KB LDS per WGP.
typedef __attribute__((ext_vector_type(2))) float v2f;
typedef __attribute__((ext_vector_type(8))) float v8f;

#define N   256   // padded window (W=256 -> Wm=255, +1 identity pad row)
#define KS  257   // LDS row stride (257 % 64 == 1 -> conflict-free columns)
#define NB  16    // cholesky block
#define NTB (N / NB)
#define HAZ 0.01f

#define WMMA_F32(a, b, c) \
  __builtin_amdgcn_wmma_f32_16x16x4_f32(false, (a), false, (b), (short)0, (c), false, false)

// ---- WMMA fragment helpers (layouts per CDNA5 ISA 7.12.2) -----------------
// A 16x4 f32: lanes0-15 M=lane K={0,1}; lanes16-31 M=lane-16 K={2,3} (negated)
__device__ __forceinline__ v2f fragA_neg(const float* Km, int row0, int kb4,
                                         int ln, int half) {
  v2f a;
  a.x = -Km[(row0 + ln) * KS + kb4 + 2 * half + 0];
  a.y = -Km[(row0 + ln) * KS + kb4 + 2 * half + 1];
  return a;
}
// B 4x16 f32: B[k][n] = L(jb,kb)[n][k]
__device__ __forceinline__ v2f fragB(const float* Km, int row0, int kb4,
                                     int ln, int half) {
  v2f b;
  b.x = Km[(row0 + ln) * KS + kb4 + 2 * half + 0];
  b.y = Km[(row0 + ln) * KS + kb4 + 2 * half + 1];
  return b;
}
// C/D 16x16 f32: VGPR q holds M=q (lanes0-15) / M=q+8 (lanes16-31), N=ln
__device__ __forceinline__ v8f loadC(const float* Km, int ri, int rj,
                                     int ln, int half) {
  v8f c;
#pragma unroll
  for (int q = 0; q < 8; ++q) c[q] = Km[(ri + q + half * 8) * KS + rj + ln];
  return c;
}
__device__ __forceinline__ void storeC(float* Km, int ri, int rj,
                                       int ln, int half, v8f c) {
#pragma unroll
  for (int q = 0; q < 8; ++q) Km[(ri + q + half * 8) * KS + rj + ln] = c[q];
}

// ---- trailing-update tile kernels: C -= L(ib,kb) * L(jb,kb)^T -------------
__device__ __forceinline__ void syrk1(float* Km, int base, int ib, int jb,
                                      int ln, int half) {
  const int ri = ib * NB, rj = jb * NB;
  v8f c = loadC(Km, ri, rj, ln, half);
#pragma unroll
  for (int kk = 0; kk < 4; ++kk) {
    const int kb4 = base + kk * 4;
    c = WMMA_F32(fragA_neg(Km, ri, kb4, ln, half),
                 fragB(Km, rj, kb4, ln, half), c);
  }
  storeC(Km, ri, rj, ln, half, c);
}

// off-diagonal 2x2 super-tile: A/B fragments reused twice (halves LDS/WMMA)
__device__ __forceinline__ void syrk2x2(float* Km, int base, int ib0, int jb0,
                                        int ln, int half) {
  const int ri0 = ib0 * NB, ri1 = ri0 + NB;
  const int rj0 = jb0 * NB, rj1 = rj0 + NB;
  v8f c00 = loadC(Km, ri0, rj0, ln, half);
  v8f c01 = loadC(Km, ri0, rj1, ln, half);
  v8f c10 = loadC(Km, ri1, rj0, ln, half);
  v8f c11 = loadC(Km, ri1, rj1, ln, half);
#pragma unroll
  for (int kk = 0; kk < 4; ++kk) {
    const int kb4 = base + kk * 4;
    const v2f a0 = fragA_neg(Km, ri0, kb4, ln, half);
    const v2f a1 = fragA_neg(Km, ri1, kb4, ln, half);
    const v2f b0 = fragB(Km, rj0, kb4, ln, half);
    const v2f b1 = fragB(Km, rj1, kb4, ln, half);
    c00 = WMMA_F32(a0, b0, c00);
    c01 = WMMA_F32(a0, b1, c01);
    c10 = WMMA_F32(a1, b0, c10);
    c11 = WMMA_F32(a1, b1, c11);
  }
  storeC(Km, ri0, rj0, ln, half, c00);
  storeC(Km, ri0, rj1, ln, half, c01);
  storeC(Km, ri1, rj0, ln, half, c10);
  storeC(Km, ri1, rj1, ln, half, c11);
}

// diagonal 2x2 super-tile: only the 3 lower sub-tiles
__device__ __forceinline__ void syrk2x2d(float* Km, int base, int ib0,
                                         int ln, int half) {
  const int ri0 = ib0 * NB, ri1 = ri0 + NB;
  v8f c00 = loadC(Km, ri0, ri0, ln, half);
  v8f c10 = loadC(Km, ri1, ri0, ln, half);
  v8f c11 = loadC(Km, ri1, ri1, ln, half);
#pragma unroll
  for (int kk = 0; kk < 4; ++kk) {
    const int kb4 = base + kk * 4;
    const v2f a0 = fragA_neg(Km, ri0, kb4, ln, half);
    const v2f a1 = fragA_neg(Km, ri1, kb4, ln, half);
    const v2f b0 = fragB(Km, ri0, kb4, ln, half);
    const v2f b1 = fragB(Km, ri1, kb4, ln, half);
    c00 = WMMA_F32(a0, b0, c00);
    c10 = WMMA_F32(a1, b0, c10);
    c11 = WMMA_F32(a1, b1, c11);
  }
  storeC(Km, ri0, ri0, ln, half, c00);
  storeC(Km, ri1, ri0, ln, half, c10);
  storeC(Km, ri1, ri1, ln, half, c11);
}

// ---------------------------------------------------------------------------
// Kernel 1: per-timestep GP predictive probability (UPM).
// One workgroup (256 thr = 8 wave32) per t. K matrix lives entirely in LDS.
// ---------------------------------------------------------------------------
__global__ void __launch_bounds__(256)
upm_kernel(const float* __restrict__ X, const float* __restrict__ Y,
           const float* __restrict__ la, const float* __restrict__ ll,
           const float* __restrict__ lnz, const int* __restrict__ wptr,
           float* __restrict__ upm, int T) {
  extern __shared__ float smem[];
  float* Km  = smem;                 // N x KS
  float* xw  = smem + N * KS;        // window x
  float* yw  = xw + N;               // window y (masked)
  float* av  = yw + N;               // rhs1: Kx  -> solve
  float* vv  = av + N;               // rhs2: Yw  -> solve
  float* red = vv + N;               // 256-wide reduction

  const int tid  = threadIdx.x;
  const int wave = tid >> 5;
  const int lane = tid & 31;
  const int ln   = lane & 15;
  const int half = lane >> 4;

  const int t = blockIdx.x + 1;                  // 1-based timestep
  const float amp   = expf(la[0]);
  const float ls2   = expf(2.0f * ll[0]);
  const float noise = expf(lnz[0]);
  const float ihl2  = 0.5f / ls2;

  int W = wptr[0]; if (W > N) W = N;
  const int Wm = W - 1;
  const int m     = (t == 1) ? 1 : min(Wm, t - 1);
  const int start = (t == 1) ? 0 : (t - 1 - m);

  __builtin_prefetch(&X[start], 0, 0);           // global_prefetch_b8
  __builtin_prefetch(&Y[start], 0, 0);

  // ---- build window vectors ----
  {
    int o  = tid;
    int id = start + o; id = max(0, min(T - 1, id));
    xw[o] = X[id];
    yw[o] = (o < m) ? Y[id] : 0.0f;
  }
  __syncthreads();
  const float xq = X[t - 1];
  const float yt = Y[t - 1];

  // ---- build K symmetric (each unordered pair computed once: ~129 expf/thr)
  {
    const int  r  = tid;
    const bool vr = r < m;
    const float xr = xw[r];
    // diagonal
    Km[r * KS + r] = vr ? (amp + noise) : 1.0f;
    // circular-distance pair sweep: o=1..127 all r; o=128 only r<128
    for (int o = 1; o <= 128; ++o) {
      if (o == 128 && r >= 128) break;
      const int c = (r + o) & (N - 1);
      float d   = xr - xw[c];
      float off = (vr && (c < m)) ? amp * expf(-d * d * ihl2) : 0.0f;
      Km[r * KS + c] = off;
      Km[c * KS + r] = off;
    }
    // RHS
    float dq = xr - xq;
    av[r] = vr ? amp * expf(-dq * dq * ihl2) : 0.0f;
    vv[r] = yw[r];
  }
  __syncthreads();

  // ---- blocked Cholesky (right-looking), trailing SYRK on f32 WMMA ----
  for (int kb = 0; kb < NTB; ++kb) {
    const int base = kb * NB;

    // diagonal 16x16 factor (row-parallel over threads 0..15)
    for (int j = 0; j < NB; ++j) {
      if (tid == j) {
        float s = Km[(base + j) * KS + base + j];
        for (int p = 0; p < j; ++p) {
          float l = Km[(base + j) * KS + base + p];
          s -= l * l;
        }
        Km[(base + j) * KS + base + j] = sqrtf(s);
      }
      __syncthreads();
      if (tid > j && tid < NB) {
        float s = Km[(base + tid) * KS + base + j];
        for (int p = 0; p < j; ++p)
          s -= Km[(base + tid) * KS + base + p] * Km[(base + j) * KS + base + p];
        Km[(base + tid) * KS + base + j] = s / Km[(base + j) * KS + base + j];
      }
      __syncthreads();
    }

    // panel TRSM: each thread owns one row below the diagonal block
    const int rstart = base + NB;
    const int nrows  = N - rstart;
    if (tid < nrows) {
      const int rr = rstart + tid;
      float xr[NB];
      for (int j = 0; j < NB; ++j) {
        float s = Km[rr * KS + base + j];
        for (int p = 0; p < j; ++p)
          s -= xr[p] * Km[(base + j) * KS + base + p];
        xr[j] = s / Km[(base + j) * KS + base + j];
      }
      for (int j = 0; j < NB; ++j) Km[rr * KS + base + j] = xr[j];
    }
    __syncthreads();

    // trailing update over 2x2 super-tiles of 16x16 tiles, distributed
    // round-robin over the 8 waves. All branches are wave-uniform =>
    // EXEC is all-ones at every WMMA (ISA requirement).
    const int t0 = kb + 1;
    const int nt = NTB - t0;
    if (nt > 0) {
      const int nst = (nt + 1) >> 1;           // super-tile grid size
      int p = 0;
      for (int sj = 0; sj < nst; ++sj) {
        for (int si = sj; si < nst; ++si) {
          const int mypair = p++;
          if ((mypair & 7) != wave) continue;
          const int ib0 = t0 + 2 * si;
          const int jb0 = t0 + 2 * sj;
          const bool vi1 = (ib0 + 1) < NTB;    // false only on last super-row
          if (si == sj) {                      // diagonal super-tile
            if (vi1) syrk2x2d(Km, base, ib0, ln, half);
            else     syrk1(Km, base, ib0, jb0, ln, half);
          } else if (vi1) {                    // full off-diagonal 2x2
            syrk2x2(Km, base, ib0, jb0, ln, half);
          } else {                             // clipped last super-row: 1x2
            syrk1(Km, base, ib0, jb0, ln, half);
            syrk1(Km, base, ib0, jb0 + 1, ln, half);
          }
        }
      }
    }
    __syncthreads();
  }

  // ---- two forward substitutions L a = Kx, L v = Yw ----
  for (int j = 0; j < N; ++j) {
    if (tid == j) {
      float d = Km[j * KS + j];
      av[j] /= d;
      vv[j] /= d;
    }
    __syncthreads();
    if (tid > j) {
      float l = Km[tid * KS + j];
      av[tid] -= av[j] * l;
      vv[tid] -= vv[j] * l;
    }
    __syncthreads();
  }

  // ---- mu = a.v ; var = amp - a.a ; upm = N(yt; mu, var) ----
  const float pa = av[tid], pv = vv[tid];
  red[tid] = pa * pv;
  __syncthreads();
  for (int s = 128; s > 0; s >>= 1) {
    if (tid < s) red[tid] += red[tid + s];
    __syncthreads();
  }
  const float mu = red[0];
  __syncthreads();
  red[tid] = pa * pa;
  __syncthreads();
  for (int s = 128; s > 0; s >>= 1) {
    if (tid < s) red[tid] += red[tid + s];
    __syncthreads();
  }
  if (tid == 0) {
    float var  = amp - red[0];
    float diff = yt - mu;
    upm[t - 1] = expf(-0.5f * (logf(6.283185307179586f * var) +
                               diff * diff / var));
  }
}

// ---------------------------------------------------------------------------
// Kernel 2: sequential BOCPD run-length recursion (single workgroup).
// Uses Z == sum(w): loss += sum(w), r_new = [H, (1-H)*w[:-1]] / sum(w).
// ---------------------------------------------------------------------------
__global__ void __launch_bounds__(256)
scan_kernel(const float* __restrict__ upm, float* __restrict__ out, int T) {
  constexpr int TP1 = 2049;          // T <= 2048
  __shared__ float ra[TP1];
  __shared__ float rb[TP1];
  __shared__ float red[256];
  __shared__ float loss_s;

  const int tid = threadIdx.x;
  for (int i = tid; i < TP1; i += 256) {
    ra[i] = (i == 0) ? 1.0f : 0.0f;
    rb[i] = 0.0f;
  }
  if (tid == 0) loss_s = 0.0f;
  __syncthreads();

  float* cur = ra;
  float* nxt = rb;
  for (int t = 1; t <= T; ++t) {
    const float u = upm[t - 1];
    float part = 0.0f;
    for (int i = tid; i < t; i += 256) part += cur[i] * u;
    red[tid] = part;
    __syncthreads();
    for (int s = 128; s > 0; s >>= 1) {
      if (tid < s) red[tid] += red[tid + s];
      __syncthreads();
    }
    const float sumw = red[0];
    const float inv  = 1.0f / sumw;
    for (int i = tid; i <= T; i += 256) {
      if (i == 0) {
        nxt[0] = HAZ;                          // H*sumw / Z, Z == sumw
      } else {
        float w = (i - 1 < t) ? cur[i - 1] * u : 0.0f;
        nxt[i] = (1.0f - HAZ) * w * inv;
      }
    }
    if (tid == 0) loss_s += sumw;
    __syncthreads();
    float* tmp = cur; cur = nxt; nxt = tmp;
  }
  if (tid == 0) out[0] = loss_s;
}

// ---------------------------------------------------------------------------
extern "C" void kernel_launch(void* const* d_in, const int* in_sizes, int n_in,
                              void* d_out, int out_size, void* d_ws, size_t ws_size,
                              hipStream_t stream) {
  const float* X   = (const float*)d_in[0];
  const float* Y   = (const float*)d_in[1];
  const float* la  = (const float*)d_in[2];
  const float* ll  = (const float*)d_in[3];
  const float* lnz = (const float*)d_in[4];
  const int*   wpt = (const int*)d_in[5];
  const int T = in_sizes[1];                   // Y has T elements

  float* upm = (float*)d_ws;                   // T floats of scratch
  float* out = (float*)d_out;

  const size_t shmem = (size_t)(N * KS + 5 * N) * sizeof(float); // ~268 KB < 320 KB
  upm_kernel<<<dim3(T), dim3(256), shmem, stream>>>(X, Y, la, ll, lnz, wpt, upm, T);
  scan_kernel<<<dim3(1), dim3(256), 0, stream>>>(upm, out, T);
}